// SPDSExtractor_40106404610845
// MI455X (gfx1250) — compile-verified
//
#include <hip/hip_runtime.h>
#include <hip/hip_bf16.h>
#include <math.h>

// SPDS extractor for MI455X (gfx1250, wave32).
//
// Strategy:
//  - Phase 1 (per scale p in {2,4,8}): compute patch mean/std (+det for p=2)
//    ONLY at the 64x64 grid points the 32x32 bilinear resize can touch
//    (the {lo,hi} source indices per output row/col). -> Feat[b][ch][64][64].
//  - Phase 2: exact bilinear resize as two f32 WMMA GEMMs per (b,ch):
//        T   = Feat(64x64) @ Wx^T(64x32)      (V_WMMA_F32_16X16X4_F32)
//        Out = Wy(32x64)   @ T(64x32)
//    with Wy/Wx the banded interp-weight matrices synthesized in registers.
//
// Workspace usage: 32*19*4096 floats = ~10 MB of d_ws.

typedef __attribute__((ext_vector_type(2))) float v2f;
typedef __attribute__((ext_vector_type(8))) float v8f;

// ---- bilinear coordinate helpers (match F.interpolate align_corners=False) --
__device__ __forceinline__ float bilin_src(int o, int n) {
    float src = (o + 0.5f) * ((float)n / 32.0f) - 0.5f;
    return fminf(fmaxf(src, 0.0f), (float)(n - 1));
}
// u in [0,64): even u -> lo index of output coord u/2, odd u -> hi index.
__device__ __forceinline__ int bilin_idx(int u, int n) {
    int o = u >> 1;
    float src = bilin_src(o, n);
    int lo = (int)floorf(src);
    int hi = min(lo + 1, n - 1);
    return (u & 1) ? hi : lo;
}
__device__ __forceinline__ float bilin_w(int o, int n) {
    float src = bilin_src(o, n);
    return src - floorf(src);
}

// ---------------- Phase 1: gathered patch statistics ------------------------
// grid = (16, B), block = 256; thread t covers point (i,j) = (t/64, t%64).
template <int P, int S, int N, int NCH>
__global__ __launch_bounds__(256) void spds_phase1(const float* __restrict__ x,
                                                   float* __restrict__ feat) {
    const int b = blockIdx.y;
    const int t = blockIdx.x * 256 + threadIdx.x;   // 0..4095
    const int i = t >> 6;
    const int j = t & 63;

    const int gy = bilin_idx(i, N);                  // grid row  -> pixel row gy*S
    const int gx = bilin_idx(j, N);                  // grid col  -> pixel col gx*S
    const float inv_np = 1.0f / (float)(P * P);
    const float inv_nm1 = 1.0f / (float)(P * P - 1);

    const float* xb = x + (size_t)b * 3u * 256u * 256u;
    float det_acc = 0.0f;

#pragma unroll
    for (int c = 0; c < 3; ++c) {
        const float* xc = xb + c * 65536 + (gy * S) * 256 + gx * S;
        float s = 0.0f, s2 = 0.0f;
#pragma unroll
        for (int dy = 0; dy < P; ++dy) {
#pragma unroll
            for (int dx = 0; dx < P; ++dx) {
                float v = xc[dy * 256 + dx];
                s += v;
                s2 = fmaf(v, v, s2);
            }
        }
        float mean = s * inv_np;
        float var = (s2 - s * s * inv_np) * inv_nm1;
        float sd = sqrtf(fmaxf(var, 0.0f)) + 1e-6f;
        feat[(((size_t)b * NCH + c) * 64 + i) * 64 + j] = mean;
        feat[(((size_t)b * NCH + 3 + c) * 64 + i) * 64 + j] = sd;

        if constexpr (P == 2) {
            float a = xc[0], bb = xc[1], cc = xc[256], dd = xc[257];
            float raw = a * dd - bb * cc;
            float mag = powf(fabsf(raw), 0.7f);
            det_acc += (raw > 0.0f) ? mag : ((raw < 0.0f) ? -mag : 0.0f);
        }
    }
    if constexpr (P == 2) {
        feat[(((size_t)b * NCH + 6) * 64 + i) * 64 + j] = det_acc * (1.0f / 3.0f);
    }
}

// ---------------- Phase 2: bilinear resize via f32 WMMA ---------------------
// grid = (NCH, B), block = 32 (one wave). EXEC all-ones around every WMMA.
template <int N, int NCH, int CH_OFF>
__global__ __launch_bounds__(32) void spds_phase2(const float* __restrict__ feat,
                                                  float* __restrict__ out) {
    __shared__ float sF[64 * 68];   // Feat tile, stride 68: 16B-aligned rows,
                                    // conflict-free A-operand column reads
    __shared__ float sT[64 * 34];   // intermediate T = Feat @ Wx^T (64x32)

    const int b = blockIdx.y;
    const int c = blockIdx.x;
    const int lane = threadIdx.x;
    const int hi16 = lane >> 4;     // 0: lanes 0-15, 1: lanes 16-31
    const int l15 = lane & 15;

    // ---- stage Feat[b][c] (64x64 f32, 16 KB) into LDS, float4 coalesced ----
    const float* fp = feat + (((size_t)b * NCH + c) * 4096);
#pragma unroll
    for (int k = 0; k < 32; ++k) {
        int lin = (lane + k * 32) * 4;              // float4 granule
        int i = lin >> 6, j = lin & 63;
        float4 v = *(const float4*)(fp + lin);
        *(float4*)&sF[i * 68 + j] = v;              // row base 272B -> 16B aligned
    }
    __syncthreads();

    // ---- GEMM1: T(64x32) = Feat(64x64) @ Wx^T(64x32) -----------------------
    // Wx^T[k][n] nonzero only at k = 2n (val 1-wx) and k = 2n+1 (val wx):
    // the K-window for n-tile nt is [32*nt, 32*nt+31] -> 8 K-steps of 4.
    {
        const int nn_base = l15;                    // per-lane output column
#pragma unroll
        for (int mt = 0; mt < 4; ++mt) {
#pragma unroll
            for (int nt = 0; nt < 2; ++nt) {
                const int nn = nt * 16 + nn_base;
                const float w = bilin_w(nn, N);
                v8f acc = {};
#pragma unroll
                for (int t = 0; t < 8; ++t) {
                    const int k4 = nt * 32 + t * 4;
                    const int kb = k4 + 2 * hi16;   // this lane's K pair base
                    const int m = mt * 16 + l15;
                    // A operand: Feat[m][kb], Feat[m][kb+1]  (ds_load_b64)
                    v2f a = *(const v2f*)&sF[m * 68 + kb];
                    // B operand: Wx^T[kb][nn], Wx^T[kb+1][nn] (in registers)
                    const bool on = (kb == 2 * nn);
                    v2f bb;
                    bb.x = on ? (1.0f - w) : 0.0f;
                    bb.y = on ? w : 0.0f;
                    acc = __builtin_amdgcn_wmma_f32_16x16x4_f32(
                        false, a, false, bb, (short)0, acc, false, false);
                }
                // D layout -> LDS T: reg r holds row (mt*16 + r + 8*hi16)
#pragma unroll
                for (int r = 0; r < 8; ++r) {
                    int row = mt * 16 + r + 8 * hi16;
                    int col = nt * 16 + l15;
                    sT[row * 34 + col] = acc[r];
                }
            }
        }
    }
    __syncthreads();

    // ---- GEMM2: Out(32x32) = Wy(32x64) @ T(64x32) --------------------------
    // Wy[m][k] nonzero only at k = 2m, 2m+1 -> K-window [32*mt, 32*mt+31].
    float* op = out + (((size_t)b * 19 + CH_OFF + c) * 32) * 32;
#pragma unroll
    for (int mt = 0; mt < 2; ++mt) {
        const int m = mt * 16 + l15;                // per-lane output row
        const float wy = bilin_w(m, N);
#pragma unroll
        for (int nt = 0; nt < 2; ++nt) {
            const int n = nt * 16 + l15;
            v8f acc = {};
#pragma unroll
            for (int t = 0; t < 8; ++t) {
                const int k4 = mt * 32 + t * 4;
                const int kb = k4 + 2 * hi16;
                // A operand: Wy[m][kb], Wy[m][kb+1] (in registers)
                const bool on = (kb == 2 * m);
                v2f a;
                a.x = on ? (1.0f - wy) : 0.0f;
                a.y = on ? wy : 0.0f;
                // B operand: T[kb][n], T[kb+1][n]
                v2f bb;
                bb.x = sT[kb * 34 + n];
                bb.y = sT[(kb + 1) * 34 + n];
                acc = __builtin_amdgcn_wmma_f32_16x16x4_f32(
                    false, a, false, bb, (short)0, acc, false, false);
            }
#pragma unroll
            for (int r = 0; r < 8; ++r) {
                int oy = mt * 16 + r + 8 * hi16;
                int ox = nt * 16 + l15;
                op[oy * 32 + ox] = acc[r];
            }
        }
    }
}

extern "C" void kernel_launch(void* const* d_in, const int* in_sizes, int n_in,
                              void* d_out, int out_size, void* d_ws, size_t ws_size,
                              hipStream_t stream) {
    const float* x = (const float*)d_in[0];       // (32,3,256,256) f32
    float* out = (float*)d_out;                   // (32,19,32,32) f32
    float* ws = (float*)d_ws;

    // Workspace: gathered feature maps, 64x64 each. Total 32*19*4096*4B ~= 10 MB.
    float* feat2 = ws;                            // (32,7,64,64)  p=2
    float* feat4 = feat2 + (size_t)32 * 7 * 4096; // (32,6,64,64)  p=4
    float* feat8 = feat4 + (size_t)32 * 6 * 4096; // (32,6,64,64)  p=8

    dim3 g1(16, 32);   // 16 blocks x 256 threads = 4096 grid points, per batch
    spds_phase1<2, 1, 255, 7><<<g1, 256, 0, stream>>>(x, feat2);
    spds_phase1<4, 2, 127, 6><<<g1, 256, 0, stream>>>(x, feat4);
    spds_phase1<8, 4, 63, 6><<<g1, 256, 0, stream>>>(x, feat8);

    spds_phase2<255, 7, 0><<<dim3(7, 32), 32, 0, stream>>>(feat2, out);
    spds_phase2<127, 6, 7><<<dim3(6, 32), 32, 0, stream>>>(feat4, out);
    spds_phase2<63, 6, 13><<<dim3(6, 32), 32, 0, stream>>>(feat8, out);
}